// CoordinationHistogram_65163243815715
// MI455X (gfx1250) — compile-verified
//
#include <hip/hip_runtime.h>

// ---------------------------------------------------------------------------
// CoordinationHistogram for MI455X (gfx1250, wave32)
//
// Phase 0: zero the 8*20000-float coordination accumulator in d_ws + d_out.
// Phase 1: streaming-read bound (~128 MB => ~5.5us @ 23.3 TB/s). Each thread
//          handles 4 elements via 3x b128 nontemporal loads + 1x b128 index
//          load; LDS (80 KB of the 320 KB WGP pool) pre-aggregates the
//          switching-function scatter (ds_add_f32), cutting global f32
//          atomics from 8M to 2.56M so they stay off the critical path.
//          512 threads/block = 16 waves for HBM latency hiding.
// Phase 2: KDE bin reduction mapped onto V_WMMA_F32_16X16X4_F32:
//          A = all-ones 16x4  ->  D[m][n] = sum_k B[k][n] + C[m][n]
//          B = 4 atoms x 16 bins of exp(-2*(c-bin)^2) per WMMA issue.
// ---------------------------------------------------------------------------

#define N_ATOMS     20000
#define S_SYS       8
#define E_ELEMS     1000000
#define CHUNKS      16                     // blocks per system in phase 1
#define CHUNK_ELEMS (E_ELEMS / CHUNKS)     // 62500 (divisible by 4)
#define SC_THREADS  512                    // 16 waves/block in phase 1
#define KDE_BLKS    32                     // blocks per system in phase 2
#define R1F         4.4f                   // CUTOFF*4/5
#define INV_DR      (1.0f / 1.1f)          // 1/(R0-R1)

typedef __attribute__((ext_vector_type(2))) float v2f;
typedef __attribute__((ext_vector_type(4))) float f4;
typedef __attribute__((ext_vector_type(4))) int   i4;
typedef __attribute__((ext_vector_type(8))) float v8f;

__device__ __forceinline__ float switch_fn(float x, float y, float z) {
    const float d = sqrtf(x * x + y * y + z * z);
    const float t = (d - R1F) * INV_DR;
    const float m = (t - 1.0f) * (t - 1.0f) * (1.0f + 2.0f * t);
    return (t <= 0.0f) ? 1.0f : ((t >= 1.0f) ? 0.0f : m);
}

__global__ __launch_bounds__(256)
void ch_zero_kernel(float* __restrict__ coords, float* __restrict__ out) {
    int i = blockIdx.x * 256 + threadIdx.x;
    if (i < S_SYS * N_ATOMS) coords[i] = 0.0f;
    if (i < S_SYS * 16)      out[i]    = 0.0f;
}

__global__ __launch_bounds__(SC_THREADS)
void ch_scatter_kernel(const float* __restrict__ vec,
                       const int*   __restrict__ idx,
                       float*       __restrict__ coords) {
    __shared__ float lds[N_ATOMS];                 // 80 KB of the 320 KB WGP LDS
    const int s     = blockIdx.x / CHUNKS;
    const int chunk = blockIdx.x % CHUNKS;
    const int tid   = threadIdx.x;

    for (int a = tid; a < N_ATOMS; a += SC_THREADS) lds[a] = 0.0f;
    __syncthreads();

    const int ebase = s * E_ELEMS + chunk * CHUNK_ELEMS;
    const f4* __restrict__ vec4 = (const f4*)(vec + (size_t)ebase * 3);  // 16B aligned
    const i4* __restrict__ idx4 = (const i4*)(idx + ebase);              // 16B aligned
    const int ngroups = CHUNK_ELEMS / 4;                                 // 15625

    for (int g = tid; g < ngroups; g += SC_THREADS) {
        // 4 elements = 48 bytes of xyz data: three b128 NT loads + one b128 index load
        const f4 p0 = __builtin_nontemporal_load(&vec4[g * 3 + 0]);
        const f4 p1 = __builtin_nontemporal_load(&vec4[g * 3 + 1]);
        const f4 p2 = __builtin_nontemporal_load(&vec4[g * 3 + 2]);
        const i4 ai = __builtin_nontemporal_load(&idx4[g]);

        atomicAdd(&lds[ai.x], switch_fn(p0.x, p0.y, p0.z));   // ds_add_f32
        atomicAdd(&lds[ai.y], switch_fn(p0.w, p1.x, p1.y));
        atomicAdd(&lds[ai.z], switch_fn(p1.z, p1.w, p2.x));
        atomicAdd(&lds[ai.w], switch_fn(p2.y, p2.z, p2.w));
    }
    __syncthreads();

    float* __restrict__ outp = coords + s * N_ATOMS;
    for (int a = tid; a < N_ATOMS; a += SC_THREADS) {
        const float v = lds[a];
        if (v != 0.0f) atomicAdd(&outp[a], v);     // global_atomic_add_f32 (L2)
    }
}

__global__ __launch_bounds__(256)
void ch_kde_kernel(const float* __restrict__ coords, float* __restrict__ out) {
    const int s    = blockIdx.x / KDE_BLKS;
    const int blk  = blockIdx.x % KDE_BLKS;
    const int tid  = threadIdx.x;
    const int lane = tid & 31;
    const int wave = tid >> 5;

    const float* __restrict__ c = coords + s * N_ATOMS;

    // B-matrix lane mapping for V_WMMA_F32_16X16X4_F32 (4x16, K x N):
    //   lanes  0-15: VGPR0 = row K=0, VGPR1 = row K=1, N = lane
    //   lanes 16-31: VGPR0 = row K=2, VGPR1 = row K=3, N = lane-16
    const float fbin = (float)(lane & 15);
    const int   ksel = (lane >> 4) << 1;           // 0 or 2

    v2f a;                                          // all-ones A => D = colsum(B)+C
    a[0] = 1.0f;
    a[1] = 1.0f;
    v8f acc = {};

    // 5000 groups of 4 atoms, strided over KDE_BLKS*8 = 256 waves (wave-uniform).
    const int gstart  = blk * 8 + wave;
    const int gstride = KDE_BLKS * 8;
    for (int g = gstart; g < N_ATOMS / 4; g += gstride) {
        const int abase = g * 4 + ksel;
        const float d0 = c[abase + 0] - fbin;
        const float d1 = c[abase + 1] - fbin;
        v2f b;
        b[0] = __expf(-2.0f * d0 * d0);             // 0.5/sigma^2 = 2.0
        b[1] = __expf(-2.0f * d1 * d1);
        acc = __builtin_amdgcn_wmma_f32_16x16x4_f32(
                  /*neg_a=*/false, a, /*neg_b=*/false, b,
                  /*c_mod=*/(short)0, acc, /*reuse_a=*/false, /*reuse_b=*/false);
    }

    // acc[0] on lanes 0-15 holds D[M=0][N=lane] = this wave's 16 bin sums.
    __shared__ float partial[8][16];
    if (lane < 16) partial[wave][lane] = acc[0];
    __syncthreads();

    if (tid < 16) {
        float sum = 0.0f;
#pragma unroll
        for (int w = 0; w < 8; ++w) sum += partial[w][tid];
        atomicAdd(&out[s * 16 + tid], sum);
    }
}

extern "C" void kernel_launch(void* const* d_in, const int* in_sizes, int n_in,
                              void* d_out, int out_size, void* d_ws, size_t ws_size,
                              hipStream_t stream) {
    const float* vec = (const float*)d_in[0];   // [8, 1e6, 3] f32
    const int*   idx = (const int*)  d_in[1];   // [8, 1e6]    i32
    float*       out = (float*)d_out;           // [8, 16]     f32
    float*       ws  = (float*)d_ws;            // 8*20000 f32 accumulator

    (void)in_sizes; (void)n_in; (void)out_size; (void)ws_size;

    ch_zero_kernel<<<(S_SYS * N_ATOMS + 255) / 256, 256, 0, stream>>>(ws, out);
    ch_scatter_kernel<<<S_SYS * CHUNKS, SC_THREADS, 0, stream>>>(vec, idx, ws);
    ch_kde_kernel<<<S_SYS * KDE_BLKS, 256, 0, stream>>>(ws, out);
}